// ImplicitModalFilter_38774964748497
// MI455X (gfx1250) — compile-verified
//
#include <hip/hip_runtime.h>

// out[d, 16q+r] = sum_o (exp(glogp[d,o]*r) * R[d,o]) * exp(glogp[d,o]*16q)
// => per-d GEMM: A(16x64) x B(64x128) using V_WMMA_F32_16X16X4_F32 (exact f32).

typedef float v2f __attribute__((ext_vector_type(2)));
typedef float v8f __attribute__((ext_vector_type(8)));

#define ORDER 64
#define NCHUNK 16            // ORDER / 4 (K=4 per f32 WMMA)
#define WAVES_PER_BLOCK 8

__global__ __launch_bounds__(256)
void ImplicitModalFilter_wmma_f32(const float* __restrict__ gamma,
                                  const float* __restrict__ R,
                                  const float* __restrict__ p,
                                  float* __restrict__ out,
                                  int D, int L, int ntiles) {
    const int wave = threadIdx.x >> 5;
    const int lane = threadIdx.x & 31;
    const int d = blockIdx.x * WAVES_PER_BLOCK + wave;
    if (d >= D) return;                 // wave-uniform: EXEC all-1 for live waves

    const int rq    = lane & 15;        // row r for A, col offset q for B
    const int half  = lane >> 4;        // 0: K pair {0,1}, 1: K pair {2,3}
    const int obase = half * 2;
    const float rf  = (float)rq;

    const float* gd = gamma + (size_t)d * ORDER;
    const float* Rd = R     + (size_t)d * ORDER;
    const float* pd = p     + (size_t)d * ORDER;

    // Per-lane cached decay rates and A fragments (registers; loops fully unrolled).
    float gl0[NCHUNK], gl1[NCHUNK];
    v2f   afrag[NCHUNK];

#pragma unroll
    for (int c = 0; c < NCHUNK; ++c) {
        const int o = 4 * c + obase;
        const float g0 = gd[o], g1 = gd[o + 1];
        const float p0 = pd[o], p1 = pd[o + 1];
        const float r0 = Rd[o], r1 = Rd[o + 1];
        const float l0 = -__expf(p0) * __expf(g0);   // glogp = -exp(p)*exp(gamma)
        const float l1 = -__expf(p1) * __expf(g1);
        gl0[c] = l0;
        gl1[c] = l1;
        // A[r, o] = exp(glogp * r) * R   (16x4 f32 fragment: v2f per lane)
        afrag[c].x = __expf(l0 * rf) * r0;
        afrag[c].y = __expf(l1 * rf) * r1;
    }

    float* outd = out + (size_t)d * L;

    for (int j = 0; j < ntiles; ++j) {
        // B[o, q] = exp(glogp * 16q), q = 16j + rq  (4x16 f32 fragment: v2f per lane)
        const float q16 = 16.0f * (float)(16 * j + rq);
        v8f acc = {};
#pragma unroll
        for (int c = 0; c < NCHUNK; ++c) {
            v2f b;
            b.x = __expf(gl0[c] * q16);
            b.y = __expf(gl1[c] * q16);
            // D = A x B + C  (exact f32 matrix core path)
            acc = __builtin_amdgcn_wmma_f32_16x16x4_f32(
                /*neg_a=*/false, afrag[c],
                /*neg_b=*/false, b,
                /*c_mod=*/(short)0, acc,
                /*reuse_a=*/false, /*reuse_b=*/false);
        }
        // C/D layout: VGPR v -> row r = v + 8*half, col n = rq.
        // t = 16*(16j + n) + r  => lane owns 8 consecutive t's.
        const int tbase = 256 * j + 16 * rq + 8 * half;
        if (tbase + 7 < L) {
            float4 lo = make_float4(acc[0], acc[1], acc[2], acc[3]);
            float4 hi = make_float4(acc[4], acc[5], acc[6], acc[7]);
            *(float4*)(outd + tbase)     = lo;   // global_store_b128
            *(float4*)(outd + tbase + 4) = hi;   // global_store_b128
        }
    }
}

extern "C" void kernel_launch(void* const* d_in, const int* in_sizes, int n_in,
                              void* d_out, int out_size, void* d_ws, size_t ws_size,
                              hipStream_t stream) {
    const float* gamma = (const float*)d_in[0];
    const float* R     = (const float*)d_in[1];
    const float* p     = (const float*)d_in[2];
    float* out         = (float*)d_out;

    const int D = in_sizes[0] / ORDER;          // 2048
    const int L = (D > 0) ? (out_size / D) : 0; // 2048
    const int ntiles = L / 256;                 // 8 output tiles of 16x16 per d

    const int blocks = (D + WAVES_PER_BLOCK - 1) / WAVES_PER_BLOCK;
    ImplicitModalFilter_wmma_f32<<<blocks, 256, 0, stream>>>(gamma, R, p, out, D, L, ntiles);
}